// DecodingLoss_BCEBased_74895639707840
// MI455X (gfx1250) — compile-verified
//
#include <hip/hip_runtime.h>

#ifndef __has_builtin
#define __has_builtin(x) 0
#endif

#if __has_builtin(__builtin_amdgcn_global_load_async_to_lds_b128) && \
    __has_builtin(__builtin_amdgcn_s_wait_asynccnt)
#define USE_ASYNC_LDS 1
#else
#define USE_ASYNC_LDS 0
#endif

namespace {
constexpr int   kB       = 1024;
constexpr int   kN       = 20000;   // columns of llrs
constexpr int   kM       = 10000;   // checks
constexpr int   kK       = 8;       // observables
constexpr int   kChkW    = 8;
constexpr int   kObsW    = 200;
constexpr int   kThreads = 512;     // 16 waves (wave32)
constexpr float kEps     = 1e-6f;
constexpr float kLn2     = 0.69314718055994530942f;
} // namespace

#if USE_ASYNC_LDS
// Builtin signature (from hipcc diagnostic): param0 is
// 'int __attribute__((vector_size(16))) __device__(AS1) *', param1 is the LDS
// (AS3) counterpart, then two int immediates (offset, cpol).
typedef int v4i __attribute__((vector_size(16)));
typedef __attribute__((address_space(1))) v4i GlobalV4;
typedef __attribute__((address_space(3))) v4i LdsV4;
#endif

// BCE term: softplus(-2 atanh p) + 2 atanh(p) * y  ==  ln2 - log(1 + (1-2y) p)
__device__ __forceinline__ float bce_term(float p, float y) {
    p = fminf(fmaxf(p, -1.0f + kEps), 1.0f - kEps);
    const float s = 1.0f - 2.0f * y;     // y in {0,1} -> s in {+1,-1}, exact
    return kLn2 - logf(fmaf(s, p, 1.0f));
}

__global__ __launch_bounds__(kThreads)
void decode_loss_kernel(const float* __restrict__ llrs,
                        const float* __restrict__ syndromes,
                        const float* __restrict__ observables,
                        const int*   __restrict__ chk_cols,
                        const int*   __restrict__ obs_cols,
                        float*       __restrict__ block_out) {
    __shared__ float t_lds[kN];          // 80 KB: tanh(llrs[b,:]*0.5) table
    const int b   = blockIdx.x;
    const int tid = threadIdx.x;
    const float* row = llrs + (size_t)b * kN;

    // ---- Phase 1: stage row into LDS and apply tanh(x/2) ----
#if USE_ASYNC_LDS
    // CDNA5 async DMA: global -> LDS, 16B per lane per issue (ASYNCcnt).
    for (int i4 = tid; i4 < kN / 4; i4 += kThreads) {
        __builtin_amdgcn_global_load_async_to_lds_b128(
            (GlobalV4*)(row + 4 * i4), (LdsV4*)(t_lds + 4 * i4),
            /*offset=*/0, /*cpol=*/0);
    }
    __builtin_amdgcn_s_wait_asynccnt(0);   // this wave's copies landed
    __syncthreads();                       // all waves' copies landed
    for (int i4 = tid; i4 < kN / 4; i4 += kThreads) {
        float4 v = *(float4*)(t_lds + 4 * i4);
        v.x = tanhf(0.5f * v.x);
        v.y = tanhf(0.5f * v.y);
        v.z = tanhf(0.5f * v.z);
        v.w = tanhf(0.5f * v.w);
        *(float4*)(t_lds + 4 * i4) = v;
    }
#else
    for (int i4 = tid; i4 < kN / 4; i4 += kThreads) {
        float4 v = *(const float4*)(row + 4 * i4);
        v.x = tanhf(0.5f * v.x);
        v.y = tanhf(0.5f * v.y);
        v.z = tanhf(0.5f * v.z);
        v.w = tanhf(0.5f * v.w);
        *(float4*)(t_lds + 4 * i4) = v;   // ds_store_b128
    }
#endif
    __syncthreads();

    float acc = 0.0f;

    // ---- Phase 2: syndrome checks (dominant): 8 LDS gathers + 1 logf each ----
    for (int m = tid; m < kM; m += kThreads) {
        const int nm = m + kThreads;
        if (nm < kM) __builtin_prefetch(chk_cols + (size_t)nm * kChkW, 0, 0);
        const int4 c0 = *(const int4*)(chk_cols + (size_t)m * kChkW);
        const int4 c1 = *(const int4*)(chk_cols + (size_t)m * kChkW + 4);
        float p = t_lds[c0.x] * t_lds[c0.y] * t_lds[c0.z] * t_lds[c0.w];
        p      *= t_lds[c1.x] * t_lds[c1.y] * t_lds[c1.z] * t_lds[c1.w];
        const float y = syndromes[(size_t)b * kM + m];
        acc += 0.5f * bce_term(p, y);     // BETA = 0.5
    }

    // ---- Phase 3: observables: wave w handles obs w, lanes split 200 cols ----
    const int wave = tid >> 5;
    const int lane = tid & 31;
    if (wave < kK) {
        const int* oc = obs_cols + (size_t)wave * kObsW;
        float p = 1.0f;
        for (int j = lane; j < kObsW; j += 32) p *= t_lds[oc[j]];
        for (int off = 16; off > 0; off >>= 1) p *= __shfl_xor(p, off, 32);
        if (lane == 0) {
            const float y = observables[(size_t)b * kK + wave];
            acc += 0.5f * bce_term(p, y); // (1-BETA) = 0.5
        }
    }
    __syncthreads();                      // done reading t_lds; reuse for reduce

    // ---- Deterministic block tree-reduction ----
    t_lds[tid] = acc;
    __syncthreads();
    for (int s = kThreads >> 1; s > 0; s >>= 1) {
        if (tid < s) t_lds[tid] += t_lds[tid + s];
        __syncthreads();
    }
    if (tid == 0) block_out[b] = t_lds[0];
}

__global__ __launch_bounds__(kThreads)
void reduce_mean_kernel(const float* __restrict__ in, float* __restrict__ out) {
    __shared__ float s[kThreads];
    const int tid = threadIdx.x;
    s[tid] = in[tid] + in[tid + kThreads];   // 1024 = 2 * 512
    __syncthreads();
    for (int sh = kThreads >> 1; sh > 0; sh >>= 1) {
        if (tid < sh) s[tid] += s[tid + sh];
        __syncthreads();
    }
    if (tid == 0) out[0] = s[0] * (1.0f / (float)kB);
}

extern "C" void kernel_launch(void* const* d_in, const int* in_sizes, int n_in,
                              void* d_out, int out_size, void* d_ws, size_t ws_size,
                              hipStream_t stream) {
    (void)in_sizes; (void)n_in; (void)out_size; (void)ws_size;
    const float* llrs        = (const float*)d_in[0];
    const float* syndromes   = (const float*)d_in[1];
    const float* observables = (const float*)d_in[2];
    const int*   chk_cols    = (const int*)d_in[3];
    const int*   obs_cols    = (const int*)d_in[4];
    float* partials = (float*)d_ws;          // kB floats of scratch
    float* out      = (float*)d_out;

    decode_loss_kernel<<<dim3(kB), dim3(kThreads), 0, stream>>>(
        llrs, syndromes, observables, chk_cols, obs_cols, partials);
    reduce_mean_kernel<<<dim3(1), dim3(kThreads), 0, stream>>>(partials, out);
}